// FlowConsistLoss_4277787427354
// MI455X (gfx1250) — compile-verified
//
#include <hip/hip_runtime.h>
#include <hip/hip_bf16.h>

typedef float v2f __attribute__((ext_vector_type(2)));
typedef float v8f __attribute__((ext_vector_type(8)));

// Problem constants (fixed by setup_inputs): B=8, C=2, T=8, H=W=512.
constexpr int THREADS = 256;                  // 8 waves (wave32)
constexpr int BLOCKS  = 4096;                 // 64 blocks/image * 64 images
constexpr int HWd     = 512 * 512;            // 2^18

// Bilinear sample (zero padding) of channel planes p0/p1 at (x, y);
// returns |s0 + u0| + |s1 + u1|. Matches the reference's per-tap
// validity*weight semantics exactly.
__device__ __forceinline__ float warp_term(const float* __restrict__ p0,
                                           const float* __restrict__ p1,
                                           float x, float y,
                                           float u0, float u1) {
  float x0f = floorf(x), y0f = floorf(y);
  float fx = x - x0f, fy = y - y0f;
  float gx = 1.0f - fx, gy = 1.0f - fy;
  int xi  = (int)x0f, yi  = (int)y0f;        // exact: x0f/y0f integer-valued
  int xi1 = xi + 1,   yi1 = yi + 1;
  // One unsigned compare per axis-tap: (uint)v < 512  <=>  0 <= v <= 511.
  float wx0 = ((unsigned)xi  < 512u) ? gx : 0.0f;
  float wx1 = ((unsigned)xi1 < 512u) ? fx : 0.0f;
  float wy0 = ((unsigned)yi  < 512u) ? gy : 0.0f;
  float wy1 = ((unsigned)yi1 < 512u) ? fy : 0.0f;
  int x0 = min(max(xi,  0), 511);            // v_med3_i32
  int y0 = min(max(yi,  0), 511);
  int x1 = min(max(xi1, 0), 511);            // clamped index only matters
  int y1 = min(max(yi1, 0), 511);            // when its weight is nonzero
  float w00 = wx0 * wy0, w01 = wx1 * wy0, w10 = wx0 * wy1, w11 = wx1 * wy1;
  int r0 = y0 << 9, r1 = y1 << 9;
  int i00 = r0 + x0, i01 = r0 + x1, i10 = r1 + x0, i11 = r1 + x1;
  float s0 = fmaf(w11, p0[i11], fmaf(w10, p0[i10], fmaf(w01, p0[i01], w00 * p0[i00])));
  float s1 = fmaf(w11, p1[i11], fmaf(w10, p1[i10], fmaf(w01, p1[i01], w00 * p1[i00])));
  return fabsf(s0 + u0) + fabsf(s1 + u1);
}

__global__ __launch_bounds__(THREADS)
void fcl_main(const float* __restrict__ flow, const float* __restrict__ flowback,
              const float* __restrict__ mfw, const float* __restrict__ mbw,
              float* __restrict__ partial) {
  __shared__ float wavesum[THREADS / 32];

  // Block -> one image (all plane bases are loop-invariant SGPRs).
  const int img = blockIdx.x >> 6;            // [0,64): n = t*B + b
  const int blk = blockIdx.x & 63;            // 64 slabs of 8 rows per image
  const int t = img >> 3;
  const int b = img & 7;
  // flow/flowback plane (b,c,t): ((b*2 + c)*8 + t) * HW ; mask plane: (b*8+t)*HW
  const float* F0 = flow     + ((size_t)(b * 16 + t) << 18);
  const float* F1 = F0 + (8 << 18);
  const float* G0 = flowback + ((size_t)(b * 16 + t) << 18);
  const float* G1 = G0 + (8 << 18);
  const float* MF = mfw + ((size_t)(b * 8 + t) << 18);
  const float* MB = mbw + ((size_t)(b * 8 + t) << 18);

  const int tid = threadIdx.x;
  const int rb  = (blk << 12) + tid;          // slab base + lane
  const float wf0 = (float)tid;               // x-coord, even half-row
  const float wf1 = (float)(tid + 256);       // x-coord, odd half-row
  const float hb  = (float)(blk << 3);        // first row of this slab

  float acc = 0.0f;
  #pragma unroll 2
  for (int i = 0; i < 8; ++i) {               // 8 rows * 2 half-rows = 4096 px
    const float hf = hb + (float)i;
    const int r0 = rb + (i << 9);
    {
      float f0 = F0[r0], f1 = F1[r0], g0 = G0[r0], g1 = G1[r0];
      float tn = warp_term(G0, G1, wf0 + f0, hf + f1, f0, f1);
      float tp = warp_term(F0, F1, wf0 + g0, hf + g1, g0, g1);
      acc = fmaf(__builtin_nontemporal_load(MF + r0), tn, acc);
      acc = fmaf(__builtin_nontemporal_load(MB + r0), tp, acc);
    }
    const int r1 = r0 + 256;
    {
      float f0 = F0[r1], f1 = F1[r1], g0 = G0[r1], g1 = G1[r1];
      float tn = warp_term(G0, G1, wf1 + f0, hf + f1, f0, f1);
      float tp = warp_term(F0, F1, wf1 + g0, hf + g1, g0, g1);
      acc = fmaf(__builtin_nontemporal_load(MF + r1), tn, acc);
      acc = fmaf(__builtin_nontemporal_load(MB + r1), tp, acc);
    }
  }

  // ---- Wave32 reduction via V_WMMA_F32_16X16X4_F32 with B = ones ----
  // Each lane's accumulator sits in its first A-VGPR (second zeroed), so
  // D = A*ones holds row sums; column N=0 of D lives in lane 0 (rows 0..7,
  // VGPRs 0..7) and lane 16 (rows 8..15). lane0 + lane16 => 32-lane sum.
  v2f A; A.x = acc; A.y = 0.0f;
  v2f Bn; Bn.x = 1.0f; Bn.y = 1.0f;
  v8f Cz = {};
  v8f D = __builtin_amdgcn_wmma_f32_16x16x4_f32(
      /*neg_a=*/false, A, /*neg_b=*/false, Bn,
      /*c_mod=*/(short)0, Cz, /*reuse_a=*/false, /*reuse_b=*/false);
  float rs = ((D[0] + D[1]) + (D[2] + D[3])) + ((D[4] + D[5]) + (D[6] + D[7]));
  float wsum = __shfl(rs, 0, 32) + __shfl(rs, 16, 32);

  const int lane = tid & 31;
  const int wave = tid >> 5;
  if (lane == 0) wavesum[wave] = wsum;
  __syncthreads();
  if (tid == 0) {
    float s = 0.0f;
    #pragma unroll
    for (int i = 0; i < THREADS / 32; ++i) s += wavesum[i];
    partial[blockIdx.x] = s;
  }
}

__global__ __launch_bounds__(256)
void fcl_final(const float* __restrict__ partial, const int* __restrict__ npf,
               float* __restrict__ out) {
  __shared__ double sh[256];
  double s = 0.0;
  for (int i = threadIdx.x; i < BLOCKS; i += 256) s += (double)partial[i];
  sh[threadIdx.x] = s;
  __syncthreads();
  if (threadIdx.x == 0) {
    double tot = 0.0;
    for (int i = 0; i < 256; ++i) tot += sh[i];
    // mean denominator: [64, 2, 512, 512] elements for each loss term
    const double ntot = 64.0 * 2.0 * (double)HWd;
    out[0] = (float)(tot / ntot * (double)npf[0]);
  }
}

extern "C" void kernel_launch(void* const* d_in, const int* in_sizes, int n_in,
                              void* d_out, int out_size, void* d_ws, size_t ws_size,
                              hipStream_t stream) {
  const float* flow     = (const float*)d_in[0];
  const float* flowback = (const float*)d_in[1];
  const float* mask_fw  = (const float*)d_in[2];
  const float* mask_bw  = (const float*)d_in[3];
  const int*   npf      = (const int*)d_in[4];
  float* partial = (float*)d_ws;     // BLOCKS floats = 16 KB scratch

  fcl_main<<<BLOCKS, THREADS, 0, stream>>>(flow, flowback, mask_fw, mask_bw, partial);
  fcl_final<<<1, 256, 0, stream>>>(partial, npf, (float*)d_out);
}